// LC_est_KG_79336635892126
// MI455X (gfx1250) — compile-verified
//
#include <hip/hip_runtime.h>
#include <hip/hip_bf16.h>
#include <math.h>

#define H 2048
#define CONV_L 99

typedef __attribute__((ext_vector_type(2))) float v2f;
typedef __attribute__((ext_vector_type(8))) float v8f;

// ---------------- workspace layout (float offsets) ----------------
#define OFF_RNN   0        // 132  (rnn_in, padded)
#define OFF_GI0   256      // 6144
#define OFF_GH0   6400     // 6144
#define OFF_H1    12544    // 2048
#define OFF_GI1   14592    // 6144
#define OFF_GH1   20736    // 6144
#define OFF_A2    26880    // 2048  lrelu(h2)
#define OFF_AL2   28928    // 1024  lrelu(L2)
#define OFF_A3    29952    // 256   lrelu(L3)
#define OFF_KG    30208    // 54

// =================================================================
// Kernel 1: features + CNN -> rnn_in (129 values + zero pad)
// =================================================================
__global__ void prep_kernel(const float* __restrict__ meas,
                            const float* __restrict__ dr_temp,
                            const float* __restrict__ imu,
                            const float* __restrict__ dr,
                            const float* __restrict__ xoo,
                            const float* __restrict__ xoo_prev,
                            const float* __restrict__ xpo_prev,
                            const float* __restrict__ meas_old,
                            const float* __restrict__ w1,
                            const float* __restrict__ b1,
                            const float* __restrict__ w2,
                            const float* __restrict__ b2c,
                            float* __restrict__ rnn_in) {
  __shared__ float sx[12 * CONV_L];   // normalized dr/imu diffs, [c][j]
  __shared__ float sc[32 * CONV_L];   // conv1 output, [oc][l]
  __shared__ float cn[12];
  const int t = threadIdx.x;

  // diffs: columns 0-2 dp, 3-5 dv, 6-8 da, 9-11 dg
  for (int i = t; i < 12 * CONV_L; i += blockDim.x) {
    int c = i / CONV_L, j = i % CONV_L;
    float v;
    if (c < 6) v = dr[(j + 1) * 6 + c] - dr[j * 6 + c];
    else       v = imu[(j + 1) * 6 + (c - 6)] - imu[j * 6 + (c - 6)];
    sx[c * CONV_L + j] = v;
  }
  __syncthreads();
  if (t < 12) {
    float s = 0.f;
    for (int j = 0; j < CONV_L; ++j) { float v = sx[t * CONV_L + j]; s += v * v; }
    cn[t] = fmaxf(sqrtf(s), 1e-12f);
  }
  __syncthreads();
  for (int i = t; i < 12 * CONV_L; i += blockDim.x) sx[i] /= cn[i / CONV_L];
  __syncthreads();

  // conv1: 12 -> 32 channels, K=5, same padding, relu
  for (int i = t; i < 32 * CONV_L; i += blockDim.x) {
    int oc = i / CONV_L, l = i % CONV_L;
    float acc = b1[oc];
    for (int ic = 0; ic < 12; ++ic) {
      const float* wp = w1 + (oc * 12 + ic) * 5;
      for (int k = 0; k < 5; ++k) {
        int p = l + k - 2;
        if (p >= 0 && p < CONV_L) acc += wp[k] * sx[ic * CONV_L + p];
      }
    }
    sc[i] = acc > 0.f ? acc : 0.f;
  }
  __syncthreads();

  // conv2: 32 -> 1 channel -> rnn_in[30 + l]
  for (int l = t; l < CONV_L; l += blockDim.x) {
    float acc = b2c[0];
    for (int ic = 0; ic < 32; ++ic) {
      const float* wp = w2 + ic * 5;
      for (int k = 0; k < 5; ++k) {
        int p = l + k - 2;
        if (p >= 0 && p < CONV_L) acc += wp[k] * sc[ic * CONV_L + p];
      }
    }
    rnn_in[30 + l] = acc;
  }

  // features (30 values): trivial work, single lane
  if (t == 0) {
    float d[9], s, n;
    s = 0.f; for (int i = 0; i < 9; ++i) { d[i] = xoo[i] - xoo_prev[i]; s += d[i] * d[i]; }
    n = fmaxf(sqrtf(s), 1e-12f); for (int i = 0; i < 9; ++i) rnn_in[i] = d[i] / n;
    s = 0.f; for (int i = 0; i < 9; ++i) { d[i] = xoo[i] - xpo_prev[i]; s += d[i] * d[i]; }
    n = fmaxf(sqrtf(s), 1e-12f); for (int i = 0; i < 9; ++i) rnn_in[9 + i] = d[i] / n;
    s = 0.f; for (int i = 0; i < 6; ++i) { d[i] = meas[i] - meas_old[i]; s += d[i] * d[i]; }
    n = fmaxf(sqrtf(s), 1e-12f); for (int i = 0; i < 6; ++i) rnn_in[18 + i] = d[i] / n;
    s = 0.f; for (int i = 0; i < 6; ++i) { d[i] = dr_temp[i] - meas[i]; s += d[i] * d[i]; }
    n = fmaxf(sqrtf(s), 1e-12f); for (int i = 0; i < 6; ++i) rnn_in[24 + i] = d[i] / n;
    rnn_in[129] = 0.f; rnn_in[130] = 0.f; rnn_in[131] = 0.f;
  }
}

// =================================================================
// Kernel 2: GEMV via V_WMMA_F32_16X16X4_F32 (exact fp32 matrix path)
//   out[r] = act(W[r,:] @ x + bias[r]),  W row-major (R x K).
// One wave handles a 16-row tile. A = broadcast x (A[m][k] = x[k]);
// B[k][n] = W[row0+n][k]. Every D element in lane n == y[row0+n],
// so extraction is component 0 of the accumulator, no reduction.
// 4 independent accumulator chains (K unrolled by 16) keep ~4x b64
// W-loads in flight per lane: with only ~384 waves for the 6144x2048
// GEMVs, per-wave MLP is what sustains the 23.3 TB/s HBM stream.
// EXEC is all-1s across every WMMA (divergence is wave-uniform).
// =================================================================
__global__ void gemv_wmma(const float* __restrict__ W,
                          const float* __restrict__ x,
                          const float* __restrict__ bias,
                          float* __restrict__ out,
                          int R, int K, int act) {
  const int wave = (int)((blockIdx.x * blockDim.x + threadIdx.x) >> 5);
  const int lane = threadIdx.x & 31;
  const int row0 = wave * 16;
  if (row0 >= R) return;                 // wave-uniform
  const int n    = lane & 15;
  const int half = lane >> 4;            // K-half: 0 -> K0,K1 ; 1 -> K2,K3
  const int row  = row0 + n;
  const int rowc = row < R ? row : R - 1;  // clamp keeps EXEC full
  const float* __restrict__ Wr = W + (size_t)rowc * (size_t)K;
  const int h2 = 2 * half;

  v8f acc0 = {0.f, 0.f, 0.f, 0.f, 0.f, 0.f, 0.f, 0.f};
  v8f acc1 = {0.f, 0.f, 0.f, 0.f, 0.f, 0.f, 0.f, 0.f};
  v8f acc2 = {0.f, 0.f, 0.f, 0.f, 0.f, 0.f, 0.f, 0.f};
  v8f acc3 = {0.f, 0.f, 0.f, 0.f, 0.f, 0.f, 0.f, 0.f};

  const int K16 = K & ~15;
  for (int k0 = 0; k0 < K16; k0 += 16) {
    // Issue all loads up front: 4 independent WMMA chains.
    v2f a0, a1, a2, a3, b0, b1v, b2v, b3;
    b0.x  = Wr[k0 +  0 + h2]; b0.y  = Wr[k0 +  1 + h2];
    b1v.x = Wr[k0 +  4 + h2]; b1v.y = Wr[k0 +  5 + h2];
    b2v.x = Wr[k0 +  8 + h2]; b2v.y = Wr[k0 +  9 + h2];
    b3.x  = Wr[k0 + 12 + h2]; b3.y  = Wr[k0 + 13 + h2];
    a0.x  = x[k0 +  0 + h2];  a0.y  = x[k0 +  1 + h2];
    a1.x  = x[k0 +  4 + h2];  a1.y  = x[k0 +  5 + h2];
    a2.x  = x[k0 +  8 + h2];  a2.y  = x[k0 +  9 + h2];
    a3.x  = x[k0 + 12 + h2];  a3.y  = x[k0 + 13 + h2];
    acc0 = __builtin_amdgcn_wmma_f32_16x16x4_f32(false, a0, false, b0,
                                                 (short)0, acc0, false, false);
    acc1 = __builtin_amdgcn_wmma_f32_16x16x4_f32(false, a1, false, b1v,
                                                 (short)0, acc1, false, false);
    acc2 = __builtin_amdgcn_wmma_f32_16x16x4_f32(false, a2, false, b2v,
                                                 (short)0, acc2, false, false);
    acc3 = __builtin_amdgcn_wmma_f32_16x16x4_f32(false, a3, false, b3,
                                                 (short)0, acc3, false, false);
  }
  for (int k0 = K16; k0 + 3 < K; k0 += 4) {
    v2f a0, b0;
    a0.x = x[k0 + h2];   a0.y = x[k0 + 1 + h2];
    b0.x = Wr[k0 + h2];  b0.y = Wr[k0 + 1 + h2];
    acc0 = __builtin_amdgcn_wmma_f32_16x16x4_f32(false, a0, false, b0,
                                                 (short)0, acc0, false, false);
  }

  float y = (acc0[0] + acc1[0]) + (acc2[0] + acc3[0]);
  const int K4 = K & ~3;
  float tail = 0.f;
  for (int k = K4; k < K; ++k) tail += Wr[k] * x[k];

  if (half == 0 && row < R) {
    float v = y + tail + (bias ? bias[row] : 0.f);
    if (act) v = v > 0.f ? v : 0.01f * v;
    out[row] = v;
  }
}

// =================================================================
// Kernel 3: GRU gate combine  h = (1-z)*n + z*h_prev
// =================================================================
__global__ void gru_gate(const float* __restrict__ gi,
                         const float* __restrict__ gh,
                         const float* __restrict__ hprev,
                         float* __restrict__ hout, int apply_lrelu) {
  int i = blockIdx.x * blockDim.x + threadIdx.x;
  if (i >= H) return;
  float r  = 1.f / (1.f + __expf(-(gi[i] + gh[i])));
  float z  = 1.f / (1.f + __expf(-(gi[H + i] + gh[H + i])));
  float nn = tanhf(gi[2 * H + i] + r * gh[2 * H + i]);
  float h  = (1.f - z) * nn + z * hprev[i];
  if (apply_lrelu) h = h > 0.f ? h : 0.01f * h;
  hout[i] = h;
}

// =================================================================
// Kernel 4: bias heads + Kalman update -> d_out (33 floats)
//   out = [dr_new(18), est_C_b_e_old(9), imu_error(6)]
// =================================================================
__global__ void final_kernel(const float* __restrict__ dr_temp,
                             const float* __restrict__ meas,
                             const float* __restrict__ C,
                             const float* __restrict__ Wb1,
                             const float* __restrict__ Wb2,
                             const float* __restrict__ a3,
                             const float* __restrict__ KG,
                             float* __restrict__ out) {
  __shared__ float ab[3], gb[3];
  int t = threadIdx.x;
  if (t < 6) {
    const float* Wr = (t < 3) ? (Wb1 + t * 256) : (Wb2 + (t - 3) * 256);
    float s = 0.f;
    for (int i = 0; i < 256; ++i) s += Wr[i] * a3[i];
    if (t < 3) ab[t] = s; else gb[t - 3] = s;
  }
  __syncthreads();
  if (t == 0) {
    float sa = fabsf(ab[0]) + fabsf(ab[1]) + fabsf(ab[2]);
    float sg = fabsf(gb[0]) + fabsf(gb[1]) + fabsf(gb[2]);
    float fa = sa > 0.5f ? 0.1f : 1.f;
    float fg = sg > 0.01f ? 0.01f : 1.f;
    float dz[6];
    for (int c = 0; c < 6; ++c) dz[c] = dr_temp[c] - meas[c];
    for (int r = 0; r < 6; ++r) {
      float inno = 0.f;
      for (int c = 0; c < 6; ++c) inno += KG[r * 6 + c] * dz[c];
      out[r] = dr_temp[r] - inno;
    }
    out[6] = dr_temp[6]; out[7] = dr_temp[7]; out[8] = dr_temp[8];
    for (int i = 0; i < 9; ++i) out[9 + i]  = C[i];   // dr_new[9:18]
    for (int i = 0; i < 9; ++i) out[18 + i] = C[i];   // est_C_b_e_old
    for (int i = 0; i < 3; ++i) out[27 + i] = ab[i] * fa;
    for (int i = 0; i < 3; ++i) out[30 + i] = gb[i] * fg;
  }
}

// =================================================================
extern "C" void kernel_launch(void* const* d_in, const int* in_sizes, int n_in,
                              void* d_out, int out_size, void* d_ws, size_t ws_size,
                              hipStream_t stream) {
  const float* meas      = (const float*)d_in[0];
  const float* dr_temp   = (const float*)d_in[1];
  const float* Cbe       = (const float*)d_in[2];
  const float* imu       = (const float*)d_in[3];
  const float* dr        = (const float*)d_in[4];
  const float* h0        = (const float*)d_in[5];   // (2,1,2048)
  const float* xoo       = (const float*)d_in[6];
  const float* xoo_prev  = (const float*)d_in[7];
  const float* xpo_prev  = (const float*)d_in[8];
  const float* meas_old  = (const float*)d_in[9];
  const float* c1w       = (const float*)d_in[10];
  const float* c1b       = (const float*)d_in[11];
  const float* c2w       = (const float*)d_in[12];
  const float* c2b       = (const float*)d_in[13];
  const float* Wih0      = (const float*)d_in[14];
  const float* Whh0      = (const float*)d_in[15];
  const float* bih0      = (const float*)d_in[16];
  const float* bhh0      = (const float*)d_in[17];
  const float* Wih1      = (const float*)d_in[18];
  const float* Whh1      = (const float*)d_in[19];
  const float* bih1      = (const float*)d_in[20];
  const float* bhh1      = (const float*)d_in[21];
  const float* W2        = (const float*)d_in[22];
  const float* b2        = (const float*)d_in[23];
  const float* W3        = (const float*)d_in[24];
  const float* b3        = (const float*)d_in[25];
  const float* Wout      = (const float*)d_in[26];
  const float* bout      = (const float*)d_in[27];
  const float* Wb1       = (const float*)d_in[28];
  const float* Wb2       = (const float*)d_in[29];

  float* ws = (float*)d_ws;
  float* outp = (float*)d_out;

  prep_kernel<<<1, 256, 0, stream>>>(meas, dr_temp, imu, dr, xoo, xoo_prev,
                                     xpo_prev, meas_old, c1w, c1b, c2w, c2b,
                                     ws + OFF_RNN);

  auto gemv = [&](const float* W, const float* x, const float* b, float* o,
                  int R, int K, int act) {
    int tiles  = (R + 15) / 16;
    int blocks = (tiles + 7) / 8;   // 8 waves (256 threads) per block
    gemv_wmma<<<blocks, 256, 0, stream>>>(W, x, b, o, R, K, act);
  };

  // GRU layer 0
  gemv(Wih0, ws + OFF_RNN, bih0, ws + OFF_GI0, 3 * H, 129, 0);
  gemv(Whh0, h0,           bhh0, ws + OFF_GH0, 3 * H, H,   0);
  gru_gate<<<(H + 255) / 256, 256, 0, stream>>>(ws + OFF_GI0, ws + OFF_GH0,
                                                h0, ws + OFF_H1, 0);
  // GRU layer 1 (output fused with leaky-relu for the MLP)
  gemv(Wih1, ws + OFF_H1, bih1, ws + OFF_GI1, 3 * H, H, 0);
  gemv(Whh1, h0 + H,      bhh1, ws + OFF_GH1, 3 * H, H, 0);
  gru_gate<<<(H + 255) / 256, 256, 0, stream>>>(ws + OFF_GI1, ws + OFF_GH1,
                                                h0 + H, ws + OFF_A2, 1);
  // MLP head
  gemv(W2,   ws + OFF_A2,  b2,   ws + OFF_AL2, 1024, H,    1);
  gemv(W3,   ws + OFF_AL2, b3,   ws + OFF_A3,  256,  1024, 1);
  gemv(Wout, ws + OFF_A3,  bout, ws + OFF_KG,  54,   256,  0);

  final_kernel<<<1, 64, 0, stream>>>(dr_temp, meas, Cbe, Wb1, Wb2,
                                     ws + OFF_A3, ws + OFF_KG, outp);
}